// HierarchyInvertClassifier_82849919140121
// MI455X (gfx1250) — compile-verified
//
#include <hip/hip_runtime.h>

// ---------------------------------------------------------------------------
// HierarchyInvertClassifier: out[b,k,h,w] = sum_c Wmat[k,c]*pred[b,c,h,w] + bias[k]
// Wmat is sparse: channel c contributes only to group GID[c] with weight w[c].
// Implemented as D[16,16pix] = A[16x64] x B[64x16pix] using V_WMMA_F32_16X16X4_F32
// (native f32 matrix path on gfx1250; op is HBM-bound, ~27us floor at 23.3 TB/s).
// ---------------------------------------------------------------------------

typedef __attribute__((ext_vector_type(2))) float v2f;
typedef __attribute__((ext_vector_type(8))) float v8f;

#define HW_  (512 * 512)
#define C_   64
#define K_   12
#define B_   8

// group id per input channel (from MAPPING, insertion order)
__device__ __constant__ int d_GID[64] = {
    0,                                  // nodata  [0]
    1, 1, 1,                            // water   [1..3]
    2, 2, 2, 2, 2, 2, 2, 2, 2, 2,       // forest  [4..13]
    3, 3, 3, 3, 3, 3,                   // shrub   [14..19]
    4, 4, 4, 4, 4, 4, 4, 4,             // grass   [20..27]
    5, 5, 5, 5, 5, 5, 5, 5, 5, 5,       // crop    [28..37]
    6, 6, 6, 6, 6, 6, 6, 6,             // urban   [38..45]
    7, 7, 7, 7, 7, 7,                   // bare    [46..51]
    8, 8, 8, 8, 8,                      // wetland [52..56]
    9, 9,                               // snow    [57..58]
    10, 10, 10,                         // moss    [59..61]
    11, 11                              // cloud   [62..63]
};

__global__ __launch_bounds__(256)
void HierarchyInvertClassifier_82849919140121_kernel(
    const float* __restrict__ pred,     // [B, 64, H, W]
    const float* __restrict__ weights,  // [64]
    const float* __restrict__ biases,   // [12]
    float* __restrict__ out)            // [B, 12, H, W]
{
    const int lane = threadIdx.x & 31;
    const int half = lane >> 4;     // 0: lanes 0-15, 1: lanes 16-31
    const int l15  = lane & 15;

    const int wave   = blockIdx.x * (blockDim.x >> 5) + (threadIdx.x >> 5);
    const int nwaves = gridDim.x * (blockDim.x >> 5);

    // -------- Preload the 16x64 mixing matrix as 16 WMMA A-fragments --------
    // A-fragment (16x4 f32, 2 VGPRs): lanes 0-15 = M rows for K={0,1},
    // lanes 16-31 = same M rows for K={2,3}; .x -> K even slot, .y -> K odd.
    v2f a[16];
#pragma unroll
    for (int k = 0; k < 16; ++k) {
        const int c0 = 4 * k + 2 * half;          // channel for .x
        const float w0 = weights[c0];
        const float w1 = weights[c0 + 1];
        a[k].x = (d_GID[c0]     == l15) ? w0 : 0.0f;
        a[k].y = (d_GID[c0 + 1] == l15) ? w1 : 0.0f;
    }

    // Per-lane bias for D rows: VGPR r of D holds row m = r + 8*half.
    float bias_r[8];
#pragma unroll
    for (int r = 0; r < 8; ++r) {
        const int m = r + 8 * half;
        bias_r[r] = (m < K_) ? biases[m] : 0.0f;
    }

    const int tilesPerBatch = HW_ / 16;           // 16384
    const int totalTiles    = B_ * tilesPerBatch; // 131072

    for (int t = wave; t < totalTiles; t += nwaves) {   // wave-uniform loop
        const int b   = t / tilesPerBatch;
        const int pix = (t - b * tilesPerBatch) * 16 + l15;   // 64B-aligned tiles

        // B-fragment addressing: lane holds pred[b, c, pix] with
        // c = 4k + 2*half (+1 for .y); each b32 load = two 64B segments.
        const float* base = pred + (size_t)b * (C_ * HW_)
                                 + (size_t)(2 * half) * HW_ + pix;

        v2f bf[16];
#pragma unroll
        for (int k = 0; k < 16; ++k) {
            bf[k].x = base[(size_t)(4 * k)     * HW_];
            bf[k].y = base[(size_t)(4 * k + 1) * HW_];
        }

        // Two accumulators halve the dependent WMMA D->C chain.
        v8f acc0 = {0.f, 0.f, 0.f, 0.f, 0.f, 0.f, 0.f, 0.f};
        v8f acc1 = {0.f, 0.f, 0.f, 0.f, 0.f, 0.f, 0.f, 0.f};
#pragma unroll
        for (int k = 0; k < 16; k += 2) {
            acc0 = __builtin_amdgcn_wmma_f32_16x16x4_f32(
                false, a[k],     false, bf[k],     (short)0, acc0, false, false);
            acc1 = __builtin_amdgcn_wmma_f32_16x16x4_f32(
                false, a[k + 1], false, bf[k + 1], (short)0, acc1, false, false);
        }

        // D layout: VGPR r = row m = r + 8*half, column = pixel l15.
        // Only rows 0..11 are real classes; rows 12..15 are padding.
        float* obase = out + (size_t)b * (K_ * HW_) + pix;
#pragma unroll
        for (int r = 0; r < 8; ++r) {
            const int m = r + 8 * half;
            if (m < K_) {
                obase[(size_t)m * HW_] = acc0[r] + acc1[r] + bias_r[r];
            }
        }
    }
}

extern "C" void kernel_launch(void* const* d_in, const int* in_sizes, int n_in,
                              void* d_out, int out_size, void* d_ws, size_t ws_size,
                              hipStream_t stream) {
    const float* pred    = (const float*)d_in[0];   // [8,64,512,512] f32
    const float* weights = (const float*)d_in[1];   // [64] f32
    const float* biases  = (const float*)d_in[2];   // [12] f32
    float* out = (float*)d_out;                     // [8,12,512,512] f32

    dim3 block(256);   // 8 waves of 32
    dim3 grid(2048);   // 16384 waves -> 8 tiles per wave (grid-stride)
    hipLaunchKernelGGL(HierarchyInvertClassifier_82849919140121_kernel,
                       grid, block, 0, stream, pred, weights, biases, out);
}